// mLSTMCell_39298950758700
// MI455X (gfx1250) — compile-verified
//
#include <hip/hip_runtime.h>
#include <math.h>

typedef __attribute__((ext_vector_type(2))) float v2f;
typedef __attribute__((ext_vector_type(4))) float v4f;
typedef __attribute__((ext_vector_type(8))) float v8f;

#define EPSV 1e-5f
// D=512, H=8, U=1024, B=8, H*U=8192

// ---- workspace layout (float offsets) ----
#define WS_XL   0         // [16,1024] x_left (rows 8..15 zero-padded for WMMA)
#define WS_XR   16384     // [8,1024]  x_right
#define WS_XLN  24576     // [8,512]   layernormed x
#define WS_BERT 28672     // [8,1024]  skip + swish(x_right)
#define WS_Q    36864     // [8,8,1024]
#define WS_K    102400    // [8,8,1024]
#define WS_V    167936    // [8,8,1024]
#define WS_O    233472    // [8,8,1024]
#define WS_NUM  299008    // [8,8,1024]
#define WS_IG   364544    // [8,8]
#define WS_FG   364608    // [8,8]
#define WS_DEN  364672    // [8,8]
#define WS_GMU  364736    // [8,2]
#define WS_GRS  364752    // [8,2]
#define WS_TOT  364800    // [16,8192] total (rows 8..15 zero-padded for WMMA)
#define WS_YP   495872    // [16 chunks][8 batch][512] = 65536
// total 561408 floats (~2.2 MB)

__device__ __forceinline__ float sigm(float z) { return 1.0f / (1.0f + expf(-z)); }
__device__ __forceinline__ float swish(float z) { return z * sigm(z); }

__device__ __forceinline__ float blk_sum(float v, float* red) {
  int tid = threadIdx.x;
  red[tid] = v; __syncthreads();
  #pragma unroll
  for (int o = 128; o > 0; o >>= 1) {
    if (tid < o) red[tid] += red[tid + o];
    __syncthreads();
  }
  float r = red[0];
  __syncthreads();
  return r;
}

__device__ __forceinline__ float wave_sum(float v) {
  #pragma unroll
  for (int m = 16; m > 0; m >>= 1) v += __shfl_xor(v, m);
  return v;
}

// ---------------- 0. zero the WMMA M-pad rows (batch 8..15) ----------------
__global__ void k_zero(float* __restrict__ ws) {
  int idx = blockIdx.x * 256 + threadIdx.x;  // 73728 total
  if (idx < 8192) {
    ws[WS_XL + 8192 + idx] = 0.0f;           // x_left rows 8..15
  } else {
    ws[WS_TOT + 65536 + (idx - 8192)] = 0.0f; // total rows 8..15
  }
}

// ---------------- 1. LayerNorm ----------------
__global__ void k_ln(const float* __restrict__ x, const float* __restrict__ g,
                     const float* __restrict__ bb, float* __restrict__ ws) {
  __shared__ float red[256];
  int b = blockIdx.x, tid = threadIdx.x;
  float x0 = x[b * 512 + tid], x1 = x[b * 512 + 256 + tid];
  float mu = blk_sum(x0 + x1, red) * (1.0f / 512.0f);
  float d0 = x0 - mu, d1 = x1 - mu;
  float var = blk_sum(d0 * d0 + d1 * d1, red) * (1.0f / 512.0f);
  float rstd = rsqrtf(var + EPSV);
  ws[WS_XLN + b * 512 + tid]       = d0 * rstd * g[tid] + bb[tid];
  ws[WS_XLN + b * 512 + 256 + tid] = d1 * rstd * g[tid + 256] + bb[tid + 256];
}

// ---------------- 2. up projections x_left/x_right ----------------
__global__ void k_up(const float* __restrict__ Wl, const float* __restrict__ Wr,
                     float* __restrict__ ws) {
  __shared__ float s_xln[512];
  __shared__ float s_part[256];
  int tile = blockIdx.x, b = blockIdx.y, tid = threadIdx.x;
  s_xln[tid]       = ws[WS_XLN + b * 512 + tid];
  s_xln[tid + 256] = ws[WS_XLN + b * 512 + tid + 256];
  __syncthreads();
  int ui = tid & 127, half = tid >> 7;
  int u = tile * 128 + ui;
  float dl = 0.f, dr = 0.f;
  const float* wl = Wl + (size_t)u * 512;
  const float* wr = Wr + (size_t)u * 512;
  for (int d = half * 256; d < half * 256 + 256; ++d) {
    float xv = s_xln[d];
    dl += xv * wl[d];
    dr += xv * wr[d];
  }
  s_part[tid] = dl; __syncthreads();
  float lsum = s_part[ui] + s_part[ui + 128]; __syncthreads();
  s_part[tid] = dr; __syncthreads();
  float rsum = s_part[ui] + s_part[ui + 128];
  if (half == 0) {
    ws[WS_XL + b * 1024 + u] = lsum;
    ws[WS_XR + b * 1024 + u] = rsum;
  }
}

// ---------------- 3. conv + swish, block-diag q/k/v, i/f gates ----------------
__global__ void k_cqg(const float* __restrict__ conv_w, const float* __restrict__ conv_b,
                      const float* __restrict__ Wq, const float* __restrict__ Wk,
                      const float* __restrict__ Wv, const float* __restrict__ Wi,
                      const float* __restrict__ Wf, float* __restrict__ ws) {
  __shared__ float s_xl[1024];
  __shared__ float s_xcs[1024];
  int b = blockIdx.x, tid = threadIdx.x;
  #pragma unroll
  for (int t = 0; t < 4; ++t) s_xl[tid + 256 * t] = ws[WS_XL + b * 1024 + tid + 256 * t];
  __syncthreads();
  float w0 = conv_w[0], w1 = conv_w[1], w2 = conv_w[2], w3 = conv_w[3], cb = conv_b[0];
  #pragma unroll
  for (int t = 0; t < 4; ++t) {
    int u = tid + 256 * t;
    float xm2 = (u >= 2) ? s_xl[u - 2] : 0.f;
    float xm1 = (u >= 1) ? s_xl[u - 1] : 0.f;
    float xp1 = (u + 1 < 1024) ? s_xl[u + 1] : 0.f;
    float xc = w0 * xm2 + w1 * xm1 + w2 * s_xl[u] + w3 * xp1 + cb;
    s_xcs[u] = swish(xc);
  }
  __syncthreads();
  // q,k,v (block-diagonal 4x4 per head)
  #pragma unroll
  for (int t = 0; t < 4; ++t) {
    int u = tid + 256 * t;
    int n = u >> 2, o = u & 3;
    float c0 = s_xcs[n * 4 + 0], c1 = s_xcs[n * 4 + 1], c2 = s_xcs[n * 4 + 2], c3 = s_xcs[n * 4 + 3];
    float l0 = s_xl[n * 4 + 0],  l1 = s_xl[n * 4 + 1],  l2 = s_xl[n * 4 + 2],  l3 = s_xl[n * 4 + 3];
    for (int h = 0; h < 8; ++h) {
      const float* wq = Wq + ((size_t)(h * 256 + n) * 16) + o * 4;
      const float* wk = Wk + ((size_t)(h * 256 + n) * 16) + o * 4;
      const float* wv = Wv + ((size_t)(h * 256 + n) * 16) + o * 4;
      float qs = c0 * wq[0] + c1 * wq[1] + c2 * wq[2] + c3 * wq[3];
      float ks = c0 * wk[0] + c1 * wk[1] + c2 * wk[2] + c3 * wk[3];
      float vs = l0 * wv[0] + l1 * wv[1] + l2 * wv[2] + l3 * wv[3];
      int idx = (b * 8 + h) * 1024 + u;
      ws[WS_Q + idx] = qs;
      ws[WS_K + idx] = ks * 0.03125f;  // 1/sqrt(1024)
      ws[WS_V + idx] = vs;
    }
  }
  // i, f gates: one wave per head
  int wid = tid >> 5, lane = tid & 31;
  int h = wid;
  float di = 0.f, df = 0.f;
  for (int j = lane; j < 1024; j += 32) {
    float xv = s_xl[j];
    di += xv * Wi[h * 1024 + j];
    df += xv * Wf[h * 1024 + j];
  }
  di = wave_sum(di);
  df = wave_sum(df);
  if (lane == 0) {
    ws[WS_IG + b * 8 + h] = expf(di);
    ws[WS_FG + b * 8 + h] = sigm(df);
  }
}

// ---------------- 4. skip projection + swish(x_right) ----------------
__global__ void k_skip(const float* __restrict__ Wskip, const float* __restrict__ bskip,
                       float* __restrict__ ws) {
  __shared__ float s_xl[1024];
  __shared__ float s_part[256];
  int tile = blockIdx.x, b = blockIdx.y, tid = threadIdx.x;
  #pragma unroll
  for (int t = 0; t < 4; ++t) s_xl[tid + 256 * t] = ws[WS_XL + b * 1024 + tid + 256 * t];
  __syncthreads();
  int ui = tid & 127, half = tid >> 7;
  int u = tile * 128 + ui;
  const float* wr = Wskip + (size_t)u * 1024;
  float s = 0.f;
  for (int j = half * 512; j < half * 512 + 512; ++j) s += s_xl[j] * wr[j];
  s_part[tid] = s; __syncthreads();
  if (half == 0) {
    float tot = s_part[ui] + s_part[ui + 128];
    float xr = ws[WS_XR + b * 1024 + u];
    ws[WS_BERT + b * 1024 + u] = tot + bskip[u] + swish(xr);
  }
}

// ---------------- 5. n-state update + den ----------------
__global__ void k_ngate(const float* __restrict__ nstate, float* __restrict__ n_out,
                        float* __restrict__ ws) {
  __shared__ float red[256];
  int blk = blockIdx.x, tid = threadIdx.x;
  int b = blk >> 3, h = blk & 7;
  float f = ws[WS_FG + blk], ig = ws[WS_IG + blk];
  float acc = 0.f;
  #pragma unroll
  for (int t = 0; t < 4; ++t) {
    int j = tid + 256 * t;
    int gidx = b * 8192 + h * 1024 + j;
    int widx = blk * 1024 + j;
    float nn = f * nstate[gidx] + ig * ws[WS_K + widx];
    n_out[gidx] = nn;
    acc += nn * ws[WS_Q + widx];
  }
  float dsum = blk_sum(acc, red);
  if (tid == 0) ws[WS_DEN + blk] = fmaxf(fabsf(dsum), 1.0f);
}

// ---------------- 6. o-gate via fp32 WMMA (D = A[16xK] x Wo^T tile) ----------------
__global__ void __launch_bounds__(32) k_ogate(const float* __restrict__ Wo,
                                              float* __restrict__ ws) {
  int lane = threadIdx.x;
  int vt = blockIdx.x, h = blockIdx.y;
  int v0 = vt * 16;
  int m = lane & 15;              // A row (batch) / B col (v)
  int koff = (lane >> 4) << 1;    // K phase: lanes 0-15 -> K0/K1, lanes 16-31 -> K2/K3
  const float* arow = ws + WS_XL + m * 1024;                       // zero-padded M
  const float* brow = Wo + (size_t)h * 1024 * 1024 + (size_t)(v0 + m) * 1024;
  v8f acc = {};
  for (int kb = 0; kb < 1024; kb += 4) {
    int kk = kb + koff;
    v2f a = *(const v2f*)(arow + kk);   // global_load_b64, unconditional
    v2f bf = *(const v2f*)(brow + kk);  // global_load_b64
    acc = __builtin_amdgcn_wmma_f32_16x16x4_f32(false, a, false, bf, (short)0, acc,
                                                false, false);
  }
  if (lane < 16) {
    #pragma unroll
    for (int r = 0; r < 8; ++r) {  // D: reg r = batch r for lanes 0-15
      ws[WS_O + (r * 8 + h) * 1024 + v0 + lane] = sigm(acc[r]);
    }
  }
}

// ---------------- 7. fused C-state stream: Cn = f*C + i*v*k, num = Cn@q ----------------
__global__ void k_cell(const float* __restrict__ C_in, float* __restrict__ C_out,
                       float* __restrict__ ws) {
  __shared__ float s_k[1024];
  __shared__ float s_q[1024];
  int rt = blockIdx.x, h = blockIdx.y, b = blockIdx.z;
  int tid = threadIdx.x;
  int bh = b * 8 + h;
  #pragma unroll
  for (int t = 0; t < 4; ++t) {
    int j = tid + 256 * t;
    s_k[j] = ws[WS_K + bh * 1024 + j];
    s_q[j] = ws[WS_Q + bh * 1024 + j];
  }
  __syncthreads();
  float f = ws[WS_FG + bh], ig = ws[WS_IG + bh];
  int wid = tid >> 5, lane = tid & 31;
  #pragma unroll
  for (int t = 0; t < 4; ++t) {
    int i = rt * 32 + wid + 8 * t;
    float iv = ig * ws[WS_V + bh * 1024 + i];
    size_t row = ((size_t)b * 1024 + i) * 8192 + (size_t)h * 1024;
    const v4f* crow = (const v4f*)(C_in + row);
    v4f* orow = (v4f*)(C_out + row);
    float acc = 0.f;
    #pragma unroll
    for (int jt = 0; jt < 8; ++jt) {
      int j4 = lane + 32 * jt;
      int j = 4 * j4;
      v4f c4 = __builtin_nontemporal_load(crow + j4);
      v4f r;
      r.x = f * c4.x + iv * s_k[j + 0];
      r.y = f * c4.y + iv * s_k[j + 1];
      r.z = f * c4.z + iv * s_k[j + 2];
      r.w = f * c4.w + iv * s_k[j + 3];
      __builtin_nontemporal_store(r, orow + j4);
      acc += r.x * s_q[j + 0] + r.y * s_q[j + 1] + r.z * s_q[j + 2] + r.w * s_q[j + 3];
    }
    acc = wave_sum(acc);
    if (lane == 0) ws[WS_NUM + bh * 1024 + i] = acc;
  }
}

// ---------------- 8. GroupNorm statistics (2 groups over [u<512 | u>=512] x H) ----
__global__ void k_gstats(float* __restrict__ ws) {
  __shared__ float red[256];
  int g = blockIdx.x, b = blockIdx.y, tid = threadIdx.x;
  float sum = 0.f, sq = 0.f;
  #pragma unroll
  for (int t = 0; t < 16; ++t) {
    int e = tid + 256 * t;           // 4096 elements: (h, u-within-group)
    int h = e >> 9;
    int u = g * 512 + (e & 511);
    int idx = (b * 8 + h) * 1024 + u;
    float hv = ws[WS_O + idx] * ws[WS_NUM + idx] / ws[WS_DEN + b * 8 + h];
    sum += hv;
    sq += hv * hv;
  }
  float S = blk_sum(sum, red);
  float Q = blk_sum(sq, red);
  if (tid == 0) {
    float mu = S * (1.0f / 4096.0f);
    float var = Q * (1.0f / 4096.0f) - mu * mu;
    ws[WS_GMU + b * 2 + g] = mu;
    ws[WS_GRS + b * 2 + g] = rsqrtf(var + EPSV);
  }
}

// ---------------- 9. total = gn(h) + bert ----------------
__global__ void k_total(const float* __restrict__ gn_g, const float* __restrict__ gn_b,
                        float* __restrict__ ws) {
  int idx = blockIdx.x * 256 + threadIdx.x;  // 65536 = b*8192 + h*1024 + u
  int b = idx >> 13;
  int r = idx & 8191;
  int h = r >> 10;
  int u = r & 1023;
  int g = u >> 9;
  int oidx = (b * 8 + h) * 1024 + u;
  float hv = ws[WS_O + oidx] * ws[WS_NUM + oidx] / ws[WS_DEN + b * 8 + h];
  float gn = (hv - ws[WS_GMU + b * 2 + g]) * ws[WS_GRS + b * 2 + g] * gn_g[u] + gn_b[u];
  ws[WS_TOT + idx] = gn + ws[WS_BERT + b * 1024 + u];
}

// ---------------- 10. y partials: total @ Wd^T via fp32 WMMA, K split ----------------
__global__ void __launch_bounds__(32) k_ydens(const float* __restrict__ Wd,
                                              float* __restrict__ ws) {
  int lane = threadIdx.x;
  int nt = blockIdx.x, chunk = blockIdx.y;
  int d0 = nt * 16;
  int m = lane & 15;
  int koff = (lane >> 4) << 1;
  const float* arow = ws + WS_TOT + m * 8192;              // zero-padded M
  const float* brow = Wd + (size_t)(d0 + m) * 8192;
  v8f acc = {};
  int k0 = chunk * 512;
  for (int kb = k0; kb < k0 + 512; kb += 4) {
    int kk = kb + koff;
    v2f a = *(const v2f*)(arow + kk);
    v2f bf = *(const v2f*)(brow + kk);
    acc = __builtin_amdgcn_wmma_f32_16x16x4_f32(false, a, false, bf, (short)0, acc,
                                                false, false);
  }
  if (lane < 16) {
    #pragma unroll
    for (int r = 0; r < 8; ++r) {
      ws[WS_YP + (chunk * 8 + r) * 512 + d0 + lane] = acc[r];
    }
  }
}

// ---------------- 11. reduce partials + residual ----------------
__global__ void k_yred(const float* __restrict__ x, const float* __restrict__ ws,
                       float* __restrict__ y) {
  int idx = blockIdx.x * 256 + threadIdx.x;  // 4096 = b*512 + d
  int b = idx >> 9;
  int d = idx & 511;
  float s = x[idx];
  #pragma unroll
  for (int c = 0; c < 16; ++c) s += ws[WS_YP + (c * 8 + b) * 512 + d];
  y[idx] = s;
}

extern "C" void kernel_launch(void* const* d_in, const int* in_sizes, int n_in,
                              void* d_out, int out_size, void* d_ws, size_t ws_size,
                              hipStream_t stream) {
  const float* x      = (const float*)d_in[0];
  const float* C_in   = (const float*)d_in[1];
  const float* nstate = (const float*)d_in[2];
  const float* ln_g   = (const float*)d_in[3];
  const float* ln_b   = (const float*)d_in[4];
  const float* Wl     = (const float*)d_in[5];
  const float* Wr     = (const float*)d_in[6];
  const float* conv_w = (const float*)d_in[7];
  const float* conv_b = (const float*)d_in[8];
  const float* Wskip  = (const float*)d_in[9];
  const float* bskip  = (const float*)d_in[10];
  const float* Wq     = (const float*)d_in[11];
  const float* Wk     = (const float*)d_in[12];
  const float* Wv     = (const float*)d_in[13];
  const float* Wi     = (const float*)d_in[14];
  const float* Wf     = (const float*)d_in[15];
  const float* Wo     = (const float*)d_in[16];
  const float* gn_g   = (const float*)d_in[17];
  const float* gn_b   = (const float*)d_in[18];
  const float* Wd     = (const float*)d_in[19];

  float* ws   = (float*)d_ws;
  float* y    = (float*)d_out;
  float* Cout = y + 4096;
  float* nout = Cout + (size_t)8 * 1024 * 8192;

  k_zero <<<288,            256, 0, stream>>>(ws);
  k_ln   <<<8,              256, 0, stream>>>(x, ln_g, ln_b, ws);
  k_up   <<<dim3(8, 8),     256, 0, stream>>>(Wl, Wr, ws);
  k_cqg  <<<8,              256, 0, stream>>>(conv_w, conv_b, Wq, Wk, Wv, Wi, Wf, ws);
  k_skip <<<dim3(8, 8),     256, 0, stream>>>(Wskip, bskip, ws);
  k_ngate<<<64,             256, 0, stream>>>(nstate, nout, ws);
  k_ogate<<<dim3(64, 8),     32, 0, stream>>>(Wo, ws);
  k_cell <<<dim3(32, 8, 8), 256, 0, stream>>>(C_in, Cout, ws);
  k_gstats<<<dim3(2, 8),    256, 0, stream>>>(ws);
  k_total<<<256,            256, 0, stream>>>(gn_g, gn_b, ws);
  k_ydens<<<dim3(32, 16),    32, 0, stream>>>(Wd, ws);
  k_yred <<<16,             256, 0, stream>>>(x, ws, y);
}